// RelationalNetwork_34686155882576
// MI455X (gfx1250) — compile-verified
//
#include <hip/hip_runtime.h>
#include <math.h>

typedef __attribute__((ext_vector_type(2))) float v2f;
typedef __attribute__((ext_vector_type(4))) float v4f;
typedef __attribute__((ext_vector_type(8))) float v8f;

#define BB 64
#define NN 256
#define DD 512          // per-tensor feature dim
#define KK 1024         // D1 + D2
#define HH 256
#define NSPLIT 4
#define ROWS_PER_SPLIT (NN / NSPLIT)   // 64

// workspace layout (in floats)
#define WS_PART 0                          // NSPLIT * BB * KK  = 262144
#define WS_MEAN (NSPLIT * BB * KK)         // + BB*KK           = 65536
#define WS_G    (WS_MEAN + BB * KK)        // + BB*HH           = 16384

// ---------------------------------------------------------------------------
// Kernel 1: streaming partial column-sums of O1/O2 over a 64-row slice of N.
// Bandwidth-bound stage: 67 MB single-use read, so use
// non-temporal loads (th:TH_LOAD_NT) to stream past L2 without thrashing it.
// ---------------------------------------------------------------------------
__global__ void k_partial_mean(const float* __restrict__ O1,
                               const float* __restrict__ O2,
                               float* __restrict__ ws) {
    const int b = blockIdx.x;    // 0..63  batch
    const int t = blockIdx.y;    // 0..1   tensor select
    const int s = blockIdx.z;    // 0..3   N split
    const int c = threadIdx.x;   // 0..127 float4 column within DD

    const float* O = (t == 0) ? O1 : O2;
    const v4f* src = reinterpret_cast<const v4f*>(O + (size_t)b * NN * DD)
                     + (size_t)(s * ROWS_PER_SPLIT) * (DD / 4) + c;

    v4f acc = {0.f, 0.f, 0.f, 0.f};
#pragma unroll 8
    for (int n = 0; n < ROWS_PER_SPLIT; ++n) {
        v4f v = __builtin_nontemporal_load(&src[(size_t)n * (DD / 4)]);
        acc += v;
    }

    v4f* dst = reinterpret_cast<v4f*>(
        ws + WS_PART + (size_t)s * BB * KK + (size_t)b * KK + t * DD);
    dst[c] = acc;
}

// ---------------------------------------------------------------------------
// Kernel 2: combine the 4 partials into the mean matrix m (64 x 1024).
// ---------------------------------------------------------------------------
__global__ void k_combine(float* __restrict__ ws) {
    const int i = blockIdx.x * blockDim.x + threadIdx.x;   // 0..16383 (v4f)
    const v4f* p = reinterpret_cast<const v4f*>(ws + WS_PART);
    const int q = BB * KK / 4;
    v4f a = p[i], b = p[i + q], c = p[i + 2 * q], d = p[i + 3 * q];
    const float inv = 1.0f / (float)NN;
    v4f r = (a + b + c + d) * inv;
    reinterpret_cast<v4f*>(ws + WS_MEAN)[i] = r;
}

// ---------------------------------------------------------------------------
// Kernel 3: g = m @ Wg + bg via V_WMMA_F32_16X16X4_F32.
// 64 waves, one 16x16 output tile per wave, 256 k-steps of K=4.
// A lane layout: row = lane%16, K = 2*(lane/16) + vgpr   (ISA 7.12.2)
// B lane layout: col = lane%16, K = 2*(lane/16) + vgpr
// C/D layout:    VGPR r -> row r + 8*(lane/16), col = lane%16
// ---------------------------------------------------------------------------
__global__ void k_gemm_wmma(const float* __restrict__ m,    // ws + WS_MEAN
                            const float* __restrict__ Wg,   // (1024, 256) row-major
                            const float* __restrict__ bg,   // (256,)
                            float* __restrict__ g) {        // ws + WS_G, (64, 256)
    const int wave  = (blockIdx.x * blockDim.x + threadIdx.x) >> 5;  // 0..63
    const int lane  = threadIdx.x & 31;
    const int bTile = wave >> 4;     // 0..3   (rows of g, 16 each)
    const int hTile = wave & 15;     // 0..15  (cols of g, 16 each)
    const int row   = lane & 15;
    const int half  = lane >> 4;

    const float* mA = m  + (size_t)(bTile * 16 + row) * KK + 2 * half;
    const float* wB = Wg + (size_t)(2 * half) * HH + hTile * 16 + row;

    v8f acc = {};
    for (int kk = 0; kk < KK; kk += 4) {
        v2f a, bm;
        a[0]  = mA[kk + 0];
        a[1]  = mA[kk + 1];
        bm[0] = wB[(size_t)(kk + 0) * HH];
        bm[1] = wB[(size_t)(kk + 1) * HH];
        acc = __builtin_amdgcn_wmma_f32_16x16x4_f32(
            /*neg_a=*/false, a, /*neg_b=*/false, bm,
            /*c_mod=*/(short)0, acc, /*reuse_a=*/false, /*reuse_b=*/false);
    }

    const int h = hTile * 16 + row;
    const float bias = bg[h];
#pragma unroll
    for (int r = 0; r < 8; ++r) {
        const int mrow = bTile * 16 + r + 8 * half;
        g[(size_t)mrow * HH + h] = acc[r] + bias;
    }
}

// ---------------------------------------------------------------------------
// Kernel 4: out[b] = sigmoid(dot(g[b,:], Wf) + bf)
// ---------------------------------------------------------------------------
__global__ void k_final(const float* __restrict__ g,
                        const float* __restrict__ Wf,
                        const float* __restrict__ bf,
                        float* __restrict__ out) {
    __shared__ float red[256];
    const int b = blockIdx.x;
    const int h = threadIdx.x;
    red[h] = g[(size_t)b * HH + h] * Wf[h];
    __syncthreads();
#pragma unroll
    for (int off = 128; off > 0; off >>= 1) {
        if (h < off) red[h] += red[h + off];
        __syncthreads();
    }
    if (h == 0) {
        const float z = red[0] + bf[0];
        out[b] = 1.0f / (1.0f + expf(-z));
    }
}

// ---------------------------------------------------------------------------
extern "C" void kernel_launch(void* const* d_in, const int* in_sizes, int n_in,
                              void* d_out, int out_size, void* d_ws, size_t ws_size,
                              hipStream_t stream) {
    const float* O1 = (const float*)d_in[0];
    const float* O2 = (const float*)d_in[1];
    const float* Wg = (const float*)d_in[2];
    const float* bg = (const float*)d_in[3];
    const float* Wf = (const float*)d_in[4];
    const float* bf = (const float*)d_in[5];
    float* out = (float*)d_out;
    float* ws  = (float*)d_ws;

    dim3 g1(BB, 2, NSPLIT);
    k_partial_mean<<<g1, 128, 0, stream>>>(O1, O2, ws);
    k_combine<<<64, 256, 0, stream>>>(ws);
    k_gemm_wmma<<<8, 256, 0, stream>>>(ws + WS_MEAN, Wg, bg, ws + WS_G);
    k_final<<<BB, 256, 0, stream>>>(ws + WS_G, Wf, bf, out);
}